// DGCN_2319282340574
// MI455X (gfx1250) — compile-verified
//
#include <hip/hip_runtime.h>
#include <math.h>

// ---------------------------------------------------------------------------
// DGCN pipeline for MI455X (gfx1250, wave32):
//   GCN(16->64) -> ReLU -> GCN(64->64) -> ReLU -> LSTM(64->128, T=16) ->
//   FC(128->64)+ReLU -> FC(64->3) -> softmax
// All GEMMs use V_WMMA_F32_16X16X4_F32 (full fp32 matrix pipe).
// B operands are pre-packed K-pair-interleaved so each lane's B fragment is a
// single aligned 8-byte (b64) coalesced load. The LSTM keeps the recurrent
// weight fragments pinned in VGPRs across all 16 timesteps (no spills).
// ---------------------------------------------------------------------------

typedef __attribute__((ext_vector_type(2))) float v2f;
typedef __attribute__((ext_vector_type(8))) float v8f;

#define FEAT      16
#define HID       64
#define GDIM      512   // 4*LSTM_DIM
#define LSTM_DIM  128
#define SEQ_LEN   16
#define SEQ_PER_BLK 16

__device__ __forceinline__ float sigmoidf_(float x) {
  return 1.0f / (1.0f + __expf(-x));
}

// One fp32 WMMA step: D(16x16) += A(16x4) * B(4x16)
__device__ __forceinline__ v8f wmma_f32_step(v2f a, v2f b, v8f c) {
  return __builtin_amdgcn_wmma_f32_16x16x4_f32(
      /*neg_a=*/false, a, /*neg_b=*/false, b,
      /*c_mod=*/(short)0, c, /*reuse_a=*/false, /*reuse_b=*/false);
}

// ---------------------------------------------------------------------------
// Generic WMMA fp32 GEMM with packed B:
//   C[M,N] = A[M,K] * B[K,N] (+bias, +relu)
// Bp layout: Bp[((k/2)*N + n)*2 + (k&1)]  -> lane B fragment = one b64 load.
// One wave computes one 16x16 tile. M,N multiples of 16; K multiple of 4.
// ---------------------------------------------------------------------------
template <int K, bool BIAS, bool RELU>
__global__ __launch_bounds__(256) void gemm_wmma_f32(
    const float* __restrict__ A, const float* __restrict__ Bp,
    const float* __restrict__ bias, float* __restrict__ C, int M, int N) {
  const int wid  = (blockIdx.x * blockDim.x + threadIdx.x) >> 5;
  const int lane = threadIdx.x & 31;
  const int ntiles = N >> 4;
  const int mtile = wid / ntiles;
  const int ntile = wid - mtile * ntiles;
  if (mtile * 16 >= M) return;  // uniform per wave -> EXEC stays all-ones

  const int nl    = lane & 15;        // M index for A frag / N index for B,C
  const int half  = lane >> 4;        // which K pair of the 4-wide step
  const int khalf = half * 2;
  const int mrow  = mtile * 16 + nl;
  const int ncol  = ntile * 16 + nl;

  v8f acc;
  float bv = BIAS ? bias[ncol] : 0.0f;
#pragma unroll
  for (int i = 0; i < 8; ++i) acc[i] = bv;

  const float* __restrict__ Ar = A + (size_t)mrow * K + khalf;
#pragma unroll 4
  for (int k0 = 0; k0 < K; k0 += 4) {
    v2f a = *(const v2f*)(Ar + k0);  // A[mrow][k0+khalf .. +1], 8B aligned
    v2f b = *(const v2f*)(Bp + ((size_t)(k0 / 2 + half) * N + ncol) * 2);
    acc = wmma_f32_step(a, b, acc);
  }

  const int rbase = mtile * 16 + half * 8;
#pragma unroll
  for (int v = 0; v < 8; ++v) {
    float o = acc[v];
    if (RELU) o = fmaxf(o, 0.0f);
    C[(size_t)(rbase + v) * N + ncol] = o;
  }
}

// ---------------------------------------------------------------------------
// Degree / normalization kernels
// ---------------------------------------------------------------------------
__global__ void k_deg_init(float* __restrict__ deg, int n) {
  int i = blockIdx.x * blockDim.x + threadIdx.x;
  if (i < n) deg[i] = 1.0f;  // + 1.0 self term
}

__global__ void k_deg_accum(const int* __restrict__ col,
                            const float* __restrict__ ew,
                            float* __restrict__ deg, int nE) {
  int stride = gridDim.x * blockDim.x;
  for (int e = blockIdx.x * blockDim.x + threadIdx.x; e < nE; e += stride)
    atomicAdd(&deg[col[e]], ew[e]);
}

__global__ void k_rsqrt_inplace(float* __restrict__ deg, int n) {
  int i = blockIdx.x * blockDim.x + threadIdx.x;
  if (i < n) deg[i] = rsqrtf(deg[i]);
}

__global__ void k_zero(float* __restrict__ p, int n) {
  int stride = gridDim.x * blockDim.x;
  for (int i = blockIdx.x * blockDim.x + threadIdx.x; i < n; i += stride)
    p[i] = 0.0f;
}

// ---------------------------------------------------------------------------
// Edge aggregation: agg[col] += (dinv[row]*ew*dinv[col]) * lin[row]  (64 feats)
// One wave per edge, each lane carries a float2 of features. agg (16 MB) and
// lin (16 MB) are L2-resident, so the f32 atomics RMW in L2.
// ---------------------------------------------------------------------------
__global__ void k_edge_agg(const float* __restrict__ lin,
                           const float* __restrict__ dinv,
                           const float* __restrict__ ew,
                           const int* __restrict__ row,
                           const int* __restrict__ col,
                           float* __restrict__ agg, int nE) {
  const int lane  = threadIdx.x & 31;
  const int wid   = (blockIdx.x * blockDim.x + threadIdx.x) >> 5;
  const int nwave = (gridDim.x * blockDim.x) >> 5;
  for (int e = wid; e < nE; e += nwave) {
    const int r = row[e];
    const int c = col[e];
    const float coef = dinv[r] * ew[e] * dinv[c];
    const float2 v = *(const float2*)(lin + (size_t)r * HID + lane * 2);
    float* dst = agg + (size_t)c * HID + lane * 2;
    atomicAdd(dst + 0, coef * v.x);
    atomicAdd(dst + 1, coef * v.y);
  }
}

// out = relu(agg + lin * dinv^2 + b)
__global__ void k_gcn_finalize(const float* __restrict__ agg,
                               const float* __restrict__ lin,
                               const float* __restrict__ dinv,
                               const float* __restrict__ b,
                               float* __restrict__ out, int nTot) {
  int i = blockIdx.x * blockDim.x + threadIdx.x;
  if (i >= nTot) return;
  int n = i >> 6, f = i & 63;
  float d = dinv[n];
  out[i] = fmaxf(agg[i] + lin[i] * d * d + b[f], 0.0f);
}

// ---------------------------------------------------------------------------
// Weight pack into K-pair-interleaved layouts:
//   K-major (K x N) sources:  W1(16x64), W2(64x64), fw1(128x64)
//       Bp[((k/2)*N + n)*2 + (k&1)] = B[k*N + n]
//   N-major sources (transpose): W_ih(512x64), W_hh(512x128)
//       Bp[((k/2)*512 + n)*2 + (k&1)] = W[n*K + k]
//   plus bsum = b_ih + b_hh
// ---------------------------------------------------------------------------
#define SZ_W1  (FEAT * HID)          // 1024
#define SZ_W2  (HID * HID)           // 4096
#define SZ_FW1 (LSTM_DIM * HID)      // 8192
#define SZ_WIH (GDIM * HID)          // 32768
#define SZ_WHH (GDIM * LSTM_DIM)     // 65536

__global__ void k_pack_weights(const float* __restrict__ W1,
                               const float* __restrict__ W2,
                               const float* __restrict__ fw1,
                               const float* __restrict__ W_ih,
                               const float* __restrict__ W_hh,
                               const float* __restrict__ b_ih,
                               const float* __restrict__ b_hh,
                               float* __restrict__ W1P,
                               float* __restrict__ W2P,
                               float* __restrict__ fw1P,
                               float* __restrict__ WihP,
                               float* __restrict__ WhhP,
                               float* __restrict__ bsum) {
  int i = blockIdx.x * blockDim.x + threadIdx.x;
  if (i < SZ_W1) {
    int k = i / HID, n = i - k * HID;
    W1P[((k >> 1) * HID + n) * 2 + (k & 1)] = W1[i];
  } else if ((i -= SZ_W1) < SZ_W2) {
    int k = i / HID, n = i - k * HID;
    W2P[((k >> 1) * HID + n) * 2 + (k & 1)] = W2[i];
  } else if ((i -= SZ_W2) < SZ_FW1) {
    int k = i / HID, n = i - k * HID;
    fw1P[((k >> 1) * HID + n) * 2 + (k & 1)] = fw1[i];
  } else if ((i -= SZ_FW1) < SZ_WIH) {
    int n = i / HID, k = i - n * HID;
    WihP[((k >> 1) * GDIM + n) * 2 + (k & 1)] = W_ih[i];
  } else if ((i -= SZ_WIH) < SZ_WHH) {
    int n = i / LSTM_DIM, k = i - n * LSTM_DIM;
    WhhP[((k >> 1) * GDIM + n) * 2 + (k & 1)] = W_hh[i];
  } else if ((i -= SZ_WHH) < GDIM) {
    bsum[i] = b_ih[i] + b_hh[i];
  }
}

// ---------------------------------------------------------------------------
// LSTM: 16 sequences per block, 1024 threads = 32 waves, ONE 16-wide gate
// tile per wave (32 tiles = 512 gates). The W_hh^T fragments for a wave's
// tile (32 x v2f = 64 VGPRs) are loaded ONCE and pinned in registers across
// all 16 timesteps; the x-part weights stream from L2 as coalesced b64 loads.
// A operands come from LDS (x_t staged, h recurrent). Gates go to LDS for the
// elementwise cell update.
// ---------------------------------------------------------------------------
__global__ __launch_bounds__(1024, 1) void k_lstm(
    const float* __restrict__ X,      // 64000 x 64 node-major
    const float* __restrict__ WihP,   // packed 32 x 512 x 2
    const float* __restrict__ WhhP,   // packed 64 x 512 x 2
    const float* __restrict__ bsum,   // 512
    float* __restrict__ Hout) {       // 4000 x 128
  __shared__ float xt[SEQ_PER_BLK * HID];        //  4 KB
  __shared__ float hbuf[SEQ_PER_BLK * LSTM_DIM]; //  8 KB
  __shared__ float cbuf[SEQ_PER_BLK * LSTM_DIM]; //  8 KB
  __shared__ float gbuf[SEQ_PER_BLK * GDIM];     // 32 KB

  const int tid  = threadIdx.x;
  const int lane = tid & 31;
  const int wave = tid >> 5;          // 0..31 == gate N-tile
  const int seqbase = blockIdx.x * SEQ_PER_BLK;
  const int nl   = lane & 15;
  const int half = lane >> 4;
  const int ncol = wave * 16 + nl;    // this lane's gate column

  // Pin recurrent-weight fragments in registers for the whole kernel.
  v2f bhh[32];
#pragma unroll
  for (int kp = 0; kp < 32; ++kp)
    bhh[kp] = *(const v2f*)(WhhP + ((size_t)(kp * 2 + half) * GDIM + ncol) * 2);

  const float bv = bsum[ncol];

  for (int i = tid; i < SEQ_PER_BLK * LSTM_DIM; i += 1024) {
    hbuf[i] = 0.0f;
    cbuf[i] = 0.0f;
  }
  __syncthreads();

  for (int t = 0; t < SEQ_LEN; ++t) {
    // stage x_t (16 seqs x 64 feats), coalesced 64-float rows
    for (int i = tid; i < SEQ_PER_BLK * HID; i += 1024) {
      int m = i >> 6, k = i & 63;
      xt[i] = X[((size_t)(seqbase + m) * SEQ_LEN + t) * HID + k];
    }
    __syncthreads();

    v8f acc;
#pragma unroll
    for (int i = 0; i < 8; ++i) acc[i] = bv;

    // x_t @ W_ih^T  (K = 64): A from LDS, B streamed from L2 (one b64 each)
#pragma unroll 4
    for (int kp = 0; kp < 16; ++kp) {
      v2f a = *(const v2f*)(xt + nl * HID + kp * 4 + half * 2);
      v2f b = *(const v2f*)(WihP + ((size_t)(kp * 2 + half) * GDIM + ncol) * 2);
      acc = wmma_f32_step(a, b, acc);
    }
    // h @ W_hh^T  (K = 128): A from LDS, B already in registers
#pragma unroll
    for (int kp = 0; kp < 32; ++kp) {
      v2f a = *(const v2f*)(hbuf + nl * LSTM_DIM + kp * 4 + half * 2);
      acc = wmma_f32_step(a, bhh[kp], acc);
    }

    // scatter this wave's 16x16 gate tile:  row m = v + 8*half
#pragma unroll
    for (int v = 0; v < 8; ++v) gbuf[(half * 8 + v) * GDIM + ncol] = acc[v];
    __syncthreads();

    // cell update: i,f,g,o splits of the 512-wide gate vector (2 elems/thread)
    for (int i = tid; i < SEQ_PER_BLK * LSTM_DIM; i += 1024) {
      int m = i >> 7, j = i & 127;
      const float* g = gbuf + m * GDIM;
      float ig = sigmoidf_(g[j]);
      float fg = sigmoidf_(g[LSTM_DIM + j]);
      float gg = tanhf(g[2 * LSTM_DIM + j]);
      float og = sigmoidf_(g[3 * LSTM_DIM + j]);
      float c = fg * cbuf[i] + ig * gg;
      cbuf[i] = c;
      hbuf[i] = og * tanhf(c);
    }
    __syncthreads();
  }

  for (int i = tid; i < SEQ_PER_BLK * LSTM_DIM; i += 1024) {
    int m = i >> 7, j = i & 127;
    Hout[(size_t)(seqbase + m) * LSTM_DIM + j] = hbuf[i];
  }
}

// ---------------------------------------------------------------------------
// Head: logits = h1 @ fw2 + fb2 ; softmax over 3 classes. One thread per row.
// ---------------------------------------------------------------------------
__global__ void k_head(const float* __restrict__ H1,   // 4000 x 64
                       const float* __restrict__ fw2,  // 64 x 3
                       const float* __restrict__ fb2,  // 3
                       float* __restrict__ out, int rows) {
  int r = blockIdx.x * blockDim.x + threadIdx.x;
  if (r >= rows) return;
  float l0 = fb2[0], l1 = fb2[1], l2 = fb2[2];
  const float* h = H1 + (size_t)r * 64;
#pragma unroll 8
  for (int k = 0; k < 64; ++k) {
    float v = h[k];
    l0 += v * fw2[k * 3 + 0];
    l1 += v * fw2[k * 3 + 1];
    l2 += v * fw2[k * 3 + 2];
  }
  float mx = fmaxf(l0, fmaxf(l1, l2));
  float e0 = __expf(l0 - mx), e1 = __expf(l1 - mx), e2 = __expf(l2 - mx);
  float inv = 1.0f / (e0 + e1 + e2);
  out[r * 3 + 0] = e0 * inv;
  out[r * 3 + 1] = e1 * inv;
  out[r * 3 + 2] = e2 * inv;
}

// ---------------------------------------------------------------------------
extern "C" void kernel_launch(void* const* d_in, const int* in_sizes, int n_in,
                              void* d_out, int out_size, void* d_ws,
                              size_t ws_size, hipStream_t stream) {
  const float* x      = (const float*)d_in[0];
  const float* eattr  = (const float*)d_in[1];   // ew (column 0 of (E,1))
  const float* W1     = (const float*)d_in[2];
  const float* b1     = (const float*)d_in[3];
  const float* W2     = (const float*)d_in[4];
  const float* b2     = (const float*)d_in[5];
  const float* W_ih   = (const float*)d_in[6];
  const float* W_hh   = (const float*)d_in[7];
  const float* b_ih   = (const float*)d_in[8];
  const float* b_hh   = (const float*)d_in[9];
  const float* fw1    = (const float*)d_in[10];
  const float* fb1    = (const float*)d_in[11];
  const float* fw2    = (const float*)d_in[12];
  const float* fb2    = (const float*)d_in[13];
  const int*   eidx   = (const int*)d_in[14];
  float*       out    = (float*)d_out;

  const int nNodes = in_sizes[0] / FEAT;   // 64000
  const int nE     = in_sizes[1];          // 1,024,000
  const int nSeq   = nNodes / SEQ_LEN;     // 4000
  const int nFeatTot = nNodes * HID;       // 4,096,000

  const int* row = eidx;        // edge_index[0]
  const int* col = eidx + nE;   // edge_index[1]

  // ---- workspace layout (floats) ----
  float* ws    = (float*)d_ws;
  float* dinv  = ws;                       // 64000 (deg, then rsqrt in place)
  float* bufA  = dinv + nNodes;            // 64000*64  (lin)
  float* bufB  = bufA + nFeatTot;          // 64000*64  (agg)
  float* bufC  = bufB + nFeatTot;          // 64000*64  (layer output)
  float* W1P   = bufC + nFeatTot;          // 1024
  float* W2P   = W1P + SZ_W1;              // 4096
  float* fw1P  = W2P + SZ_W2;              // 8192
  float* WihP  = fw1P + SZ_FW1;            // 32768
  float* WhhP  = WihP + SZ_WIH;            // 65536
  float* bsum  = WhhP + SZ_WHH;            // 512
  float* hout  = bsum + GDIM;              // 4000*128
  float* h1    = hout + nSeq * LSTM_DIM;   // 4000*64

  const int T = 256;

  // pack all GEMM B operands + bias sum
  {
    int n = SZ_W1 + SZ_W2 + SZ_FW1 + SZ_WIH + SZ_WHH + GDIM;
    k_pack_weights<<<(n + T - 1) / T, T, 0, stream>>>(
        W1, W2, fw1, W_ih, W_hh, b_ih, b_hh, W1P, W2P, fw1P, WihP, WhhP, bsum);
  }

  // degree -> dinv
  k_deg_init<<<(nNodes + T - 1) / T, T, 0, stream>>>(dinv, nNodes);
  k_deg_accum<<<1024, T, 0, stream>>>(col, eattr, dinv, nE);
  k_rsqrt_inplace<<<(nNodes + T - 1) / T, T, 0, stream>>>(dinv, nNodes);

  // ---- GCN layer 1 ----
  {
    int waves = (nNodes / 16) * (HID / 16);  // 16000
    gemm_wmma_f32<FEAT, false, false>
        <<<(waves + 7) / 8, T, 0, stream>>>(x, W1P, nullptr, bufA, nNodes, HID);
  }
  k_zero<<<4096, T, 0, stream>>>(bufB, nFeatTot);
  k_edge_agg<<<2048, T, 0, stream>>>(bufA, dinv, eattr, row, col, bufB, nE);
  k_gcn_finalize<<<(nFeatTot + T - 1) / T, T, 0, stream>>>(bufB, bufA, dinv, b1,
                                                           bufC, nFeatTot);

  // ---- GCN layer 2 ----
  {
    int waves = (nNodes / 16) * (HID / 16);
    gemm_wmma_f32<HID, false, false>
        <<<(waves + 7) / 8, T, 0, stream>>>(bufC, W2P, nullptr, bufA, nNodes, HID);
  }
  k_zero<<<4096, T, 0, stream>>>(bufB, nFeatTot);
  k_edge_agg<<<2048, T, 0, stream>>>(bufA, dinv, eattr, row, col, bufB, nE);
  k_gcn_finalize<<<(nFeatTot + T - 1) / T, T, 0, stream>>>(bufB, bufA, dinv, b2,
                                                           bufC, nFeatTot);

  // ---- LSTM: 16 seqs/block, 32 waves, recurrent weights register-resident --
  k_lstm<<<nSeq / SEQ_PER_BLK, 1024, 0, stream>>>(bufC, WihP, WhhP, bsum, hout);

  // ---- FC1 (bias+relu fused into WMMA epilogue) ----
  {
    int waves = (nSeq / 16) * (HID / 16);  // 1000
    gemm_wmma_f32<LSTM_DIM, true, true>
        <<<(waves + 7) / 8, T, 0, stream>>>(hout, fw1P, fb1, h1, nSeq, HID);
  }

  // ---- FC2 + softmax ----
  k_head<<<(nSeq + T - 1) / T, T, 0, stream>>>(h1, fw2, fb2, out, nSeq);
}